// SparseMatrixEquivariantLayer_56186762166419
// MI455X (gfx1250) — compile-verified
//
#include <hip/hip_runtime.h>

typedef float v2f __attribute__((ext_vector_type(2)));
typedef float v8f __attribute__((ext_vector_type(8)));

#define APAD 36   // padded fp32 row stride (multiple of 4 -> conflict-free b64 frag reads)
#define WPAD 36
#define TW   8    // tiles (waves) per 256-thread block

__device__ __forceinline__ v8f wmma4(v2f a, v2f b, v8f c) {
  // D = A(16x4 f32) * B(4x16 f32) + C(16x16 f32)
  return __builtin_amdgcn_wmma_f32_16x16x4_f32(false, a, false, b, (short)0, c, false, false);
}
__device__ __forceinline__ size_t minz(size_t a, size_t b) { return a < b ? a : b; }

// ws layout (float units): [sum_r n*32][sum_c n*32][diag n*32][P_row n*32][P_col n*32][P_diag n*32]
//                          [sum_all 32][diag_sum 32][gconst 32][gdiag 32]

__global__ void k_zero(float* __restrict__ ws, const int* __restrict__ nPtr) {
  size_t n = (size_t)nPtr[0];
  size_t a = 3 * n * 32;
  size_t total = a + 128;
  size_t stride = (size_t)gridDim.x * blockDim.x;
  for (size_t i = (size_t)blockIdx.x * blockDim.x + threadIdx.x; i < total; i += stride) {
    size_t idx = (i < a) ? i : (6 * n * 32 + (i - a));
    ws[idx] = 0.0f;
  }
}

__global__ void k_scatter(const float* __restrict__ values, const int* __restrict__ rows,
                          const int* __restrict__ cols, float* __restrict__ ws,
                          const int* __restrict__ nPtr, int nnz) {
  size_t n32 = (size_t)nPtr[0] * 32;
  float* sum_r = ws;
  float* sum_c = ws + n32;
  float* diag  = ws + 2 * n32;
  size_t gid = (size_t)blockIdx.x * blockDim.x + threadIdx.x;
  if (gid >= (size_t)nnz * 32) return;
  int k = (int)(gid >> 5);
  int d = (int)(gid & 31);
  float v = values[gid];
  int r = rows[k], c = cols[k];
  atomicAdd(&sum_r[(size_t)r * 32 + d], v);
  atomicAdd(&sum_c[(size_t)c * 32 + d], v);
  if (r == c) atomicAdd(&diag[(size_t)r * 32 + d], v);
}

__global__ void __launch_bounds__(256) k_reduce(float* __restrict__ ws, const int* __restrict__ nPtr) {
  int n = nPtr[0];
  size_t n32 = (size_t)n * 32;
  const float* sum_r = ws;
  const float* diag  = ws + 2 * n32;
  float* sum_all  = ws + 6 * n32;
  float* diag_sum = ws + 6 * n32 + 32;
  __shared__ float red[256];
  int d = threadIdx.x & 31;
  int g8 = threadIdx.x >> 5;   // 0..7 row slots per block
  size_t rowStride = (size_t)gridDim.x * 8;
  float sa = 0.0f, sd = 0.0f;
  for (size_t i = (size_t)blockIdx.x * 8 + g8; i < (size_t)n; i += rowStride) {
    sa += sum_r[i * 32 + d];
    sd += diag[i * 32 + d];
  }
  red[threadIdx.x] = sa;
  __syncthreads();
  if (threadIdx.x < 32) {
    float t = 0.0f;
    #pragma unroll
    for (int g = 0; g < 8; ++g) t += red[g * 32 + threadIdx.x];
    atomicAdd(&sum_all[threadIdx.x], t);
  }
  __syncthreads();
  red[threadIdx.x] = sd;
  __syncthreads();
  if (threadIdx.x < 32) {
    float t = 0.0f;
    #pragma unroll
    for (int g = 0; g < 8; ++g) t += red[g * 32 + threadIdx.x];
    atomicAdd(&diag_sum[threadIdx.x], t);
  }
}

__global__ void k_globals(const float* __restrict__ W, const float* __restrict__ bias,
                          float* __restrict__ ws, const int* __restrict__ nPtr) {
  size_t n32 = (size_t)nPtr[0] * 32;
  const float* sum_all  = ws + 6 * n32;
  const float* diag_sum = ws + 6 * n32 + 32;
  float* gconst = ws + 6 * n32 + 64;
  float* gdiag  = ws + 6 * n32 + 96;
  int d = threadIdx.x;
  if (d >= 32) return;
  float bs = 0.0f;
  for (int i = 0; i < 15; ++i) bs += bias[i];
  const float* W11 = W + 11 * 1024;
  const float* W12 = W + 12 * 1024;
  const float* W13 = W + 13 * 1024;
  const float* W14 = W + 14 * 1024;
  float gc = bs, gd = 0.0f;
  for (int k = 0; k < 32; ++k) {
    float ds = diag_sum[k], sa = sum_all[k];
    gd += ds * W11[k * 32 + d] + sa * W13[k * 32 + d];
    gc += ds * W12[k * 32 + d] + sa * W14[k * 32 + d];
  }
  gconst[d] = gc;
  gdiag[d]  = gd;
}

// P_row = sum_r@W2 + sum_c@W4 + diag@W6 ; P_col = (W3,W5,W7) ; P_diag = (W8,W9,W10)
__global__ void __launch_bounds__(256) k_nodegemm(const float* __restrict__ W,
                                                  float* __restrict__ ws,
                                                  const int* __restrict__ nPtr) {
  __shared__ float sW[9 * 32 * WPAD];      // transposed W2..W10: slot wi holds op (wi+2)
  __shared__ float sIn[TW][16 * APAD];

  const int n = nPtr[0];
  const size_t n32 = (size_t)n * 32;
  const float* src0 = ws;
  const float* src1 = ws + n32;
  const float* src2 = ws + 2 * n32;
  float* dst0 = ws + 3 * n32;
  float* dst1 = ws + 4 * n32;
  float* dst2 = ws + 5 * n32;

  const int wv = threadIdx.x >> 5;
  const int lane = threadIdx.x & 31;
  const int half = lane >> 4;
  const int lr = lane & 15;

  for (int idx = threadIdx.x; idx < 9 * 1024; idx += 256) {
    int wi = idx >> 10, rem = idx & 1023;
    int k = rem >> 5, nn = rem & 31;
    sW[wi * 32 * WPAD + nn * WPAD + k] = W[(size_t)(wi + 2) * 1024 + k * 32 + nn];
  }
  __syncthreads();

  const int ntiles = (n + 15) / 16;
  const int ngroups = (ntiles + TW - 1) / TW;
  const size_t lastNode = (size_t)(n - 1);

  for (int g = blockIdx.x; g < ngroups; g += gridDim.x) {
    const size_t i0 = ((size_t)g * TW + wv) * 16;
    v8f a00 = {}, a01 = {}, a10 = {}, a11 = {}, a20 = {}, a21 = {};
    const float* srcs[3] = {src0, src1, src2};
    #pragma unroll
    for (int s = 0; s < 3; ++s) {
      __syncthreads();
      const float* sp = srcs[s];
      #pragma unroll
      for (int i = 0; i < 16; ++i) {
        size_t node = minz(i0 + i, lastNode);
        sIn[wv][i * APAD + lane] = sp[node * 32 + lane];
      }
      __syncthreads();
      const int sl0 = s * 2 + 0;   // P_row slots 0,2,4 (W2,W4,W6)
      const int sl1 = s * 2 + 1;   // P_col slots 1,3,5 (W3,W5,W7)
      const int sl2 = 6 + s;       // P_diag slots 6,7,8 (W8,W9,W10)
      #pragma unroll
      for (int kk = 0; kk < 8; ++kk) {
        int k0 = 4 * kk + 2 * half;
        v2f a = *(const v2f*)&sIn[wv][lr * APAD + k0];
        v2f b;
        b = *(const v2f*)&sW[sl0 * 32 * WPAD + lr * WPAD + k0];        a00 = wmma4(a, b, a00);
        b = *(const v2f*)&sW[sl0 * 32 * WPAD + (16 + lr) * WPAD + k0]; a01 = wmma4(a, b, a01);
        b = *(const v2f*)&sW[sl1 * 32 * WPAD + lr * WPAD + k0];        a10 = wmma4(a, b, a10);
        b = *(const v2f*)&sW[sl1 * 32 * WPAD + (16 + lr) * WPAD + k0]; a11 = wmma4(a, b, a11);
        b = *(const v2f*)&sW[sl2 * 32 * WPAD + lr * WPAD + k0];        a20 = wmma4(a, b, a20);
        b = *(const v2f*)&sW[sl2 * 32 * WPAD + (16 + lr) * WPAD + k0]; a21 = wmma4(a, b, a21);
      }
    }
    #pragma unroll
    for (int v = 0; v < 8; ++v) {
      size_t node = i0 + v + 8 * half;
      if (node < (size_t)n) {
        dst0[node * 32 + lr] = a00[v];  dst0[node * 32 + 16 + lr] = a01[v];
        dst1[node * 32 + lr] = a10[v];  dst1[node * 32 + 16 + lr] = a11[v];
        dst2[node * 32 + lr] = a20[v];  dst2[node * 32 + 16 + lr] = a21[v];
      }
    }
  }
}

__global__ void __launch_bounds__(256) k_main(const float* __restrict__ values,
                                              const float* __restrict__ W,
                                              const int* __restrict__ rows,
                                              const int* __restrict__ cols,
                                              const int* __restrict__ perm,
                                              float* __restrict__ ws,
                                              float* __restrict__ out,
                                              const int* __restrict__ nPtr,
                                              int nnz) {
  __shared__ float sW[2 * 32 * WPAD];   // transposed W0, W1
  __shared__ float sA[TW][16 * APAD];
  __shared__ int sIdx[TW][32];          // rows[16], cols[16]

  const size_t n32 = (size_t)nPtr[0] * 32;
  const float* P_row  = ws + 3 * n32;
  const float* P_col  = ws + 4 * n32;
  const float* P_diag = ws + 5 * n32;
  const float* gconst = ws + 6 * n32 + 64;
  const float* gdiag  = ws + 6 * n32 + 96;

  const int wv = threadIdx.x >> 5;
  const int lane = threadIdx.x & 31;
  const int half = lane >> 4;
  const int lr = lane & 15;

  for (int idx = threadIdx.x; idx < 2 * 1024; idx += 256) {
    int wi = idx >> 10, rem = idx & 1023;
    int k = rem >> 5, nn = rem & 31;
    sW[wi * 32 * WPAD + nn * WPAD + k] = W[(size_t)wi * 1024 + k * 32 + nn];
  }

  const size_t e0 = ((size_t)blockIdx.x * TW + wv) * 16;
  const size_t last = (size_t)(nnz - 1);

  if (lane < 16) {
    size_t e = minz(e0 + lane, last);
    sIdx[wv][lane] = rows[e];
    sIdx[wv][16 + lane] = cols[e];
  }
  #pragma unroll
  for (int i = 0; i < 16; ++i) {
    size_t e = minz(e0 + i, last);
    sA[wv][i * APAD + lane] = values[e * 32 + lane];
  }
  __syncthreads();

  v8f acc0 = {}, acc1 = {};
  #pragma unroll
  for (int kk = 0; kk < 8; ++kk) {
    int k0 = 4 * kk + 2 * half;
    v2f a  = *(const v2f*)&sA[wv][lr * APAD + k0];
    v2f b0 = *(const v2f*)&sW[lr * WPAD + k0];
    v2f b1 = *(const v2f*)&sW[(16 + lr) * WPAD + k0];
    acc0 = wmma4(a, b0, acc0);
    acc1 = wmma4(a, b1, acc1);
  }
  __syncthreads();

  #pragma unroll
  for (int i = 0; i < 16; ++i) {
    size_t e = minz(e0 + i, last);
    size_t p = (size_t)perm[e];
    sA[wv][i * APAD + lane] = values[p * 32 + lane];
  }
  __syncthreads();

  #pragma unroll
  for (int kk = 0; kk < 8; ++kk) {
    int k0 = 4 * kk + 2 * half;
    v2f a  = *(const v2f*)&sA[wv][lr * APAD + k0];
    v2f b0 = *(const v2f*)&sW[32 * WPAD + lr * WPAD + k0];
    v2f b1 = *(const v2f*)&sW[32 * WPAD + (16 + lr) * WPAD + k0];
    acc0 = wmma4(a, b0, acc0);
    acc1 = wmma4(a, b1, acc1);
  }

  float gc0 = gconst[lr], gc1 = gconst[16 + lr];
  float gd0 = gdiag[lr],  gd1 = gdiag[16 + lr];

  #pragma unroll
  for (int v = 0; v < 8; ++v) {
    int m = v + 8 * half;
    size_t e = e0 + m;
    if (e < (size_t)nnz) {
      size_t r = (size_t)sIdx[wv][m];
      size_t c = (size_t)sIdx[wv][16 + m];
      float o0 = acc0[v] + P_row[r * 32 + lr]      + P_col[c * 32 + lr]      + gc0;
      float o1 = acc1[v] + P_row[r * 32 + 16 + lr] + P_col[c * 32 + 16 + lr] + gc1;
      if (r == c) {
        o0 += P_diag[r * 32 + lr] + gd0;
        o1 += P_diag[r * 32 + 16 + lr] + gd1;
      }
      out[e * 32 + lr] = o0;
      out[e * 32 + 16 + lr] = o1;
    }
  }
}

extern "C" void kernel_launch(void* const* d_in, const int* in_sizes, int n_in,
                              void* d_out, int out_size, void* d_ws, size_t ws_size,
                              hipStream_t stream) {
  const float* values  = (const float*)d_in[0];
  const float* weights = (const float*)d_in[1];
  const float* bias    = (const float*)d_in[2];
  const int*   rows    = (const int*)d_in[3];
  const int*   cols    = (const int*)d_in[4];
  const int*   perm    = (const int*)d_in[5];
  const int*   nPtr    = (const int*)d_in[6];   // little-endian scalar: low 32 bits hold n
  float* out = (float*)d_out;
  float* ws  = (float*)d_ws;

  long long nnz = (long long)in_sizes[0] / 32;

  k_zero<<<4096, 256, 0, stream>>>(ws, nPtr);

  long long scat = nnz * 32;
  k_scatter<<<(unsigned)((scat + 255) / 256), 256, 0, stream>>>(values, rows, cols, ws, nPtr, (int)nnz);

  k_reduce<<<512, 256, 0, stream>>>(ws, nPtr);
  k_globals<<<1, 32, 0, stream>>>(weights, bias, ws, nPtr);
  k_nodegemm<<<2048, 256, 0, stream>>>(weights, ws, nPtr);

  unsigned mainBlocks = (unsigned)((nnz + TW * 16 - 1) / (TW * 16));
  k_main<<<mainBlocks, 256, 0, stream>>>(values, weights, rows, cols, perm, ws, out, nPtr, (int)nnz);
}